// MouseGCN_32203664786076
// MI455X (gfx1250) — compile-verified
//
#include <hip/hip_runtime.h>
#include <math.h>

#define N_NODES 50000
#define N_EDGES 640000
#define IN_DIM  128
#define HID     256

typedef float v2f __attribute__((ext_vector_type(2)));
typedef float v8f __attribute__((ext_vector_type(8)));

// ---------------------------------------------------------------------------
// Degree / normalization kernels
// ---------------------------------------------------------------------------
__global__ void deg_init(float* deg, int n) {
    int i = blockIdx.x * blockDim.x + threadIdx.x;
    if (i < n) deg[i] = 1.0f;                  // self-loop contributes 1
}

__global__ void deg_accum(const int* __restrict__ dst, float* deg, int nE) {
    int i = blockIdx.x * blockDim.x + threadIdx.x;
    if (i < nE) atomicAdd(&deg[dst[i]], 1.0f);
}

__global__ void rsqrt_inplace(float* deg, int n) {
    int i = blockIdx.x * blockDim.x + threadIdx.x;
    if (i < n) deg[i] = rsqrtf(deg[i]);        // deg > 0 always (self-loop)
}

// ---------------------------------------------------------------------------
// WMMA GEMM: out[M x 256] = A[M x K] @ W[K x 256]   (optionally + bias, tanh)
// Block: 256 threads = 8 waves. Block tile 128 rows x 64 cols.
// Wave computes a 16x64 strip via 4 accumulators.
// Weight tile staged in LDS with K-pair interleaving so each B fragment
// ({W[ka][n], W[ka+1][n]}) is one aligned ds_load_b64 into an even VGPR pair
// (kills the v_mov shuffles seen in the previous round's assembly).
// Full-precision f32 WMMA (V_WMMA_F32_16X16X4_F32).
// ---------------------------------------------------------------------------
template<int K>
__global__ __launch_bounds__(256) void gemm_wmma_f32(
    const float* __restrict__ A, const float* __restrict__ W,
    const float* __restrict__ bias, float* __restrict__ out,
    int M, int Nc, int fuse_bias_tanh)
{
    // Ws layout: [K/2][64][2] -> Ws[(k>>1)*128 + 2*n + (k&1)] = W[k][n0+n]
    __shared__ float Ws[K * 64];               // K<=256 -> 64KB of 320KB LDS
    const int tid = threadIdx.x;
    const int n0  = blockIdx.y * 64;
    const int m_block = blockIdx.x * 128;

    // Cooperative staging: coalesced float4 global reads, interleaved LDS writes
    const int nvec = K * 16;                   // float4s in the tile
    for (int i = tid; i < nvec; i += 256) {
        int k  = i >> 4;
        int c4 = i & 15;
        float4 v = ((const float4*)(W + (size_t)k * Nc + n0))[c4];
        float* wp = Ws + (k >> 1) * 128 + (k & 1);
        int n = c4 * 4;
        wp[2 * (n + 0)] = v.x;
        wp[2 * (n + 1)] = v.y;
        wp[2 * (n + 2)] = v.z;
        wp[2 * (n + 3)] = v.w;
    }
    __syncthreads();

    const int wave = tid >> 5;
    const int lane = tid & 31;
    const int half = lane >> 4;                // 0: lanes 0-15, 1: lanes 16-31
    const int l15  = lane & 15;
    const int m0   = m_block + wave * 16;

    // Clamp row address so EXEC stays all-1s through the WMMA loop.
    int mrow = m0 + l15;
    if (mrow >= M) mrow = M - 1;
    const float* arow = A + (size_t)mrow * K;
    __builtin_prefetch(arow, 0, 1);            // global_prefetch_b8 on the A row

    v8f acc0 = {}, acc1 = {}, acc2 = {}, acc3 = {};

    for (int k = 0; k < K; k += 4) {
        const int ka = k + 2 * half;           // A: lanes16-31 carry K+2,K+3 (even)
        v2f a;
        a.x = arow[ka];
        a.y = arow[ka + 1];

        // B fragments: one aligned 8-byte LDS load each
        const float* wsp = Ws + (ka >> 1) * 128;
        v2f b0v = *(const v2f*)(wsp + 2 * l15);
        v2f b1v = *(const v2f*)(wsp + 2 * (16 + l15));
        v2f b2v = *(const v2f*)(wsp + 2 * (32 + l15));
        v2f b3v = *(const v2f*)(wsp + 2 * (48 + l15));

        acc0 = __builtin_amdgcn_wmma_f32_16x16x4_f32(false, a, false, b0v, (short)0, acc0, false, false);
        acc1 = __builtin_amdgcn_wmma_f32_16x16x4_f32(false, a, false, b1v, (short)0, acc1, false, false);
        acc2 = __builtin_amdgcn_wmma_f32_16x16x4_f32(false, a, false, b2v, (short)0, acc2, false, false);
        acc3 = __builtin_amdgcn_wmma_f32_16x16x4_f32(false, a, false, b3v, (short)0, acc3, false, false);
    }

    // Store: C/D layout -> VGPR r holds row m0 + 8*half + r, col n0 + 16*j + l15
    for (int r = 0; r < 8; ++r) {
        int row = m0 + half * 8 + r;
        if (row < M) {
            float* orow = out + (size_t)row * Nc + n0;
            float v0 = acc0[r], v1 = acc1[r], v2 = acc2[r], v3 = acc3[r];
            if (fuse_bias_tanh) {
                v0 = tanhf(v0 + bias[n0 + l15]);
                v1 = tanhf(v1 + bias[n0 + 16 + l15]);
                v2 = tanhf(v2 + bias[n0 + 32 + l15]);
                v3 = tanhf(v3 + bias[n0 + 48 + l15]);
            }
            orow[l15]      = v0;
            orow[16 + l15] = v1;
            orow[32 + l15] = v2;
            orow[48 + l15] = v3;
        }
    }
}

// ---------------------------------------------------------------------------
// Aggregation kernels
// ---------------------------------------------------------------------------
// Initialize accumulator with the self-loop term: out[i] = msg[i] * dinv[i]^2
__global__ __launch_bounds__(HID) void self_loop_init(
    const float* __restrict__ msg, const float* __restrict__ dinv,
    float* __restrict__ outb)
{
    int i = blockIdx.x;
    int c = threadIdx.x;
    float dv = dinv[i];
    outb[(size_t)i * HID + c] = msg[(size_t)i * HID + c] * dv * dv;
}

// One block per edge, 256 threads = channels: out[dst] += msg[src] * norm
// (both feature matrices are ~51 MB and stay resident in the 192 MB L2, so
//  these gathers/atomics are L2-bandwidth-bound, not HBM-bound)
__global__ __launch_bounds__(HID) void edge_aggregate(
    const int* __restrict__ src, const int* __restrict__ dst,
    const float* __restrict__ dinv, const float* __restrict__ msg,
    float* __restrict__ outb)
{
    int e = blockIdx.x;
    int c = threadIdx.x;
    int s = src[e];
    int d = dst[e];
    float norm = dinv[s] * dinv[d];
    atomicAdd(&outb[(size_t)d * HID + c], msg[(size_t)s * HID + c] * norm);
}

// h = tanh(h + b), one block per row
__global__ __launch_bounds__(HID) void bias_tanh(float* __restrict__ h,
                                                 const float* __restrict__ b)
{
    size_t idx = (size_t)blockIdx.x * HID + threadIdx.x;
    h[idx] = tanhf(h[idx] + b[threadIdx.x]);
}

// ---------------------------------------------------------------------------
// Launch
// ---------------------------------------------------------------------------
extern "C" void kernel_launch(void* const* d_in, const int* in_sizes, int n_in,
                              void* d_out, int out_size, void* d_ws, size_t ws_size,
                              hipStream_t stream)
{
    const float* x   = (const float*)d_in[0];
    const float* W0  = (const float*)d_in[1];
    const float* b0  = (const float*)d_in[2];
    const float* W1  = (const float*)d_in[3];
    const float* b1  = (const float*)d_in[4];
    const float* fcw = (const float*)d_in[5];
    const float* fcb = (const float*)d_in[6];
    const int*   ei  = (const int*)d_in[7];
    const int* src = ei;
    const int* dst = ei + N_EDGES;

    float* out  = (float*)d_out;
    char*  ws   = (char*)d_ws;
    float* bufA = (float*)ws;                                    // 50000x256 f32
    float* dinv = (float*)(ws + (size_t)N_NODES * HID * sizeof(float));

    // --- degree / normalization ---
    deg_init<<<(N_NODES + 255) / 256, 256, 0, stream>>>(dinv, N_NODES);
    deg_accum<<<(N_EDGES + 255) / 256, 256, 0, stream>>>(dst, dinv, N_EDGES);
    rsqrt_inplace<<<(N_NODES + 255) / 256, 256, 0, stream>>>(dinv, N_NODES);

    dim3 gemm_grid((N_NODES + 127) / 128, HID / 64);

    // --- layer 1: h1 = tanh(agg(x @ W0) + b0) ---
    gemm_wmma_f32<IN_DIM><<<gemm_grid, 256, 0, stream>>>(x, W0, b0, bufA, N_NODES, HID, 0);
    self_loop_init<<<N_NODES, HID, 0, stream>>>(bufA, dinv, out);
    edge_aggregate<<<N_EDGES, HID, 0, stream>>>(src, dst, dinv, bufA, out);
    bias_tanh<<<N_NODES, HID, 0, stream>>>(out, b0);

    // --- layer 2: h2 = tanh(agg(h1 @ W1) + b1) ---
    gemm_wmma_f32<HID><<<gemm_grid, 256, 0, stream>>>(out, W1, b1, bufA, N_NODES, HID, 0);
    self_loop_init<<<N_NODES, HID, 0, stream>>>(bufA, dinv, out);
    edge_aggregate<<<N_EDGES, HID, 0, stream>>>(src, dst, dinv, bufA, out);
    bias_tanh<<<N_NODES, HID, 0, stream>>>(out, b1);

    // --- FC: out = tanh(h2 @ fc_w + fc_b), fused epilogue, then D2D copy ---
    gemm_wmma_f32<HID><<<gemm_grid, 256, 0, stream>>>(out, fcw, fcb, bufA, N_NODES, HID, 1);
    hipMemcpyAsync(out, bufA, (size_t)N_NODES * HID * sizeof(float),
                   hipMemcpyDeviceToDevice, stream);
}